// Involution_80401787781443
// MI455X (gfx1250) — compile-verified
//
#include <hip/hip_runtime.h>

typedef float v2f __attribute__((ext_vector_type(2)));
typedef float v4f __attribute__((ext_vector_type(4)));
typedef float v8f __attribute__((ext_vector_type(8)));

// clang's builtin prototype for the async b128 copy uses V4i (gcc vector_size)
typedef int v4i __attribute__((vector_size(16)));
typedef __attribute__((address_space(1))) v4i* gv4i_p;
typedef __attribute__((address_space(3))) v4i* lv4i_p;

#define B_    16
#define C_    256
#define H_    56
#define W_    56
#define HW_   (H_ * W_)        // 3136
#define CHW_  (C_ * HW_)
#define K_    7
#define G_    16
#define CR_   64
#define KKG_  (K_ * K_ * G_)   // 784
#define M_    (B_ * HW_)       // 50176
#define EPS_  1e-5f
#define PITCH 785              // 784+1: (785 mod 64)=17 -> conflict-free pixel stride
#define CHCH  16               // channels per involution chunk == one group
#define NCHK  (C_ / CHCH)      // 16 chunks
#define XROWS 8                // rows staged per chunk: h0-3 .. h0+4

// ---- CDNA5 async global->LDS path (guarded; falls back to sync copies) ----
#if defined(__has_builtin)
#if __has_builtin(__builtin_amdgcn_global_load_async_to_lds_b128)
#define ASYNC_LOADS 1
#endif
#endif
#if defined(ASYNC_LOADS)
#if __has_builtin(__builtin_amdgcn_s_wait_asynccnt)
#define WAIT_ASYNC() __builtin_amdgcn_s_wait_asynccnt(0)
#else
#define WAIT_ASYNC() asm volatile("s_wait_asynccnt 0x0" ::: "memory")
#endif
#else
#define WAIT_ASYNC()
#endif

// ---------------------------------------------------------------------------
// Kernel A: r_pre[m][n] = x[m,:] . w1[n,:] + b1[n]      (GEMM1, fp32 WMMA)
// 784 blocks x 128 thr; each WAVE owns one M-tile and sweeps all 4 N-tiles,
// so every A fragment (x) is loaded once and feeds 4 independent WMMA chains.
// ---------------------------------------------------------------------------
__global__ __launch_bounds__(128) void k_reduce_gemm(
    const float* __restrict__ x, const float* __restrict__ w1,
    const float* __restrict__ b1, float* __restrict__ rpre) {
  const int lane = threadIdx.x & 31;
  const int wv   = threadIdx.x >> 5;
  const int mt   = blockIdx.x * 4 + wv;    // M-tile per wave
  const int mrow = lane & 15;
  const int hi   = lane >> 4;
  const int koff = hi * 2;

  const int m    = mt * 16 + mrow;
  const int bidx = m / HW_;
  const int spat = m - bidx * HW_;
  const float* xa = x + (size_t)bidx * CHW_ + spat;

  const float* wb = w1 + mrow * C_;        // + nt*16*C_ + k

  v8f acc0, acc1, acc2, acc3;
  const float bi0 = b1[mrow];
  const float bi1 = b1[16 + mrow];
  const float bi2 = b1[32 + mrow];
  const float bi3 = b1[48 + mrow];
#pragma unroll
  for (int v = 0; v < 8; ++v) {
    acc0[v] = bi0; acc1[v] = bi1; acc2[v] = bi2; acc3[v] = bi3;
  }

#pragma unroll 4
  for (int step = 0; step < C_ / 4; ++step) {
    const int k0 = step * 4 + koff;
    v2f a;
    a[0] = xa[(size_t)k0 * HW_];
    a[1] = xa[(size_t)(k0 + 1) * HW_];
    v2f w0 = *(const v2f*)(wb + 0 * 16 * C_ + k0);
    v2f w1v = *(const v2f*)(wb + 1 * 16 * C_ + k0);
    v2f w2v = *(const v2f*)(wb + 2 * 16 * C_ + k0);
    v2f w3v = *(const v2f*)(wb + 3 * 16 * C_ + k0);
    acc0 = __builtin_amdgcn_wmma_f32_16x16x4_f32(false, a, false, w0, (short)0, acc0, false, false);
    acc1 = __builtin_amdgcn_wmma_f32_16x16x4_f32(false, a, false, w1v, (short)0, acc1, false, false);
    acc2 = __builtin_amdgcn_wmma_f32_16x16x4_f32(false, a, false, w2v, (short)0, acc2, false, false);
    acc3 = __builtin_amdgcn_wmma_f32_16x16x4_f32(false, a, false, w3v, (short)0, acc3, false, false);
  }

  // C layout: VGPR v -> M = v + 8*hi, N = mrow
  float* rp = rpre + (size_t)(mt * 16) * CR_ + mrow;
#pragma unroll
  for (int v = 0; v < 8; ++v) {
    float* row = rp + (size_t)(v + 8 * hi) * CR_;
    row[0]  = acc0[v];
    row[16] = acc1[v];
    row[32] = acc2[v];
    row[48] = acc3[v];
  }
}

// ---------------------------------------------------------------------------
// Kernel B: deterministic batch stats per CR channel -> scale/shift
// ---------------------------------------------------------------------------
__global__ __launch_bounds__(256) void k_bnstats(
    const float* __restrict__ rpre, const float* __restrict__ gamma,
    const float* __restrict__ beta, float* __restrict__ ab) {
  __shared__ float ssum[256];
  __shared__ float ssq[256];
  const int n = blockIdx.x;
  const int t = threadIdx.x;
  float s = 0.f, q = 0.f;
  for (int m = t; m < M_; m += 256) {
    const float v = rpre[(size_t)m * CR_ + n];
    s += v;
    q += v * v;
  }
  ssum[t] = s;
  ssq[t]  = q;
  __syncthreads();
  for (int off = 128; off > 0; off >>= 1) {
    if (t < off) {
      ssum[t] += ssum[t + off];
      ssq[t]  += ssq[t + off];
    }
    __syncthreads();
  }
  if (t == 0) {
    const float mean = ssum[0] * (1.0f / (float)M_);
    const float var  = ssq[0] * (1.0f / (float)M_) - mean * mean;
    const float sc   = gamma[n] * rsqrtf(var + EPS_);
    ab[2 * n]     = sc;
    ab[2 * n + 1] = beta[n] - mean * sc;
  }
}

// ---------------------------------------------------------------------------
// Kernel C: fused span-GEMM (WMMA) + involution with async-LDS x staging.
// ---------------------------------------------------------------------------
__device__ __forceinline__ void stage_chunk(const float* __restrict__ x,
                                            float* __restrict__ xdst,
                                            int b0, int c0, int h0, int tid) {
  // CHCH ch x XROWS rows x 14 b128 segments; rows outside image -> zeros.
#pragma unroll 1
  for (int q = tid; q < CHCH * XROWS * 14; q += 128) {
    const int c   = q / (XROWS * 14);
    const int rem = q - c * (XROWS * 14);
    const int i   = rem / 14;
    const int seg = rem - i * 14;
    const int y   = h0 - 3 + i;
    float* dst = xdst + (c * XROWS + i) * W_ + seg * 4;
    if ((unsigned)y < (unsigned)H_) {
      const float* src =
          x + (((size_t)b0 * C_ + (c0 + c)) * H_ + y) * W_ + seg * 4;
#if defined(ASYNC_LOADS)
      __builtin_amdgcn_global_load_async_to_lds_b128(
          (gv4i_p)src, (lv4i_p)dst, 0, 0);
#else
      *(v4f*)dst = *(const v4f*)src;
#endif
    } else {
      v4f z = {0.f, 0.f, 0.f, 0.f};
      *(v4f*)dst = z;
    }
  }
}

__global__ __launch_bounds__(128) void k_span_involution(
    const float* __restrict__ x, const float* __restrict__ rpre,
    const float* __restrict__ ab, const float* __restrict__ w2,
    const float* __restrict__ b2, float* __restrict__ out) {
  __shared__ float kerlds[16 * PITCH];          // 50.2 KB
  __shared__ float abs_[2 * CR_];               // 0.5 KB
  __shared__ float xbuf[2][CHCH * XROWS * W_];  // 2 x 28 KB, double buffered

  const int tid  = threadIdx.x;
  const int lane = tid & 31;
  const int wv   = tid >> 5;
  const int mt   = blockIdx.x;

  const int m0 = mt * 16;
  const int b0 = m0 / HW_;          // tiles never cross the batch dim (3136%16==0)
  const int s0 = m0 - b0 * HW_;
  const int h0 = s0 / W_;           // tile spans rows h0 (and maybe h0+1)

  abs_[tid] = ab[tid];              // 2*CR_ == blockDim == 128

  // Kick off chunk 0 of the x patch; it flies during the WMMA phase.
  stage_chunk(x, xbuf[0], b0, 0, h0, tid);
  __syncthreads();                  // abs_ visible for phase 1

  // ---- Phase 1: ker tile = relu(bn(r)) @ w2^T + b2  (WMMA f32 16x16x4) ----
  const int mrow = lane & 15;
  const int hi   = lane >> 4;
  const int koff = hi * 2;

  v2f afrag[CR_ / 4];
  const float* ra = rpre + (size_t)(mt * 16 + mrow) * CR_;
#pragma unroll
  for (int st = 0; st < CR_ / 4; ++st) {
    const int k0 = st * 4 + koff;
    v2f r = *(const v2f*)(ra + k0);
    afrag[st][0] = fmaxf(0.f, abs_[2 * k0] * r[0] + abs_[2 * k0 + 1]);
    afrag[st][1] = fmaxf(0.f, abs_[2 * (k0 + 1)] * r[1] + abs_[2 * (k0 + 1) + 1]);
  }

  for (int nt = wv; nt < KKG_ / 16; nt += 4) {   // 49 N-tiles over 4 waves
    const int n     = nt * 16 + mrow;
    const float* wb = w2 + n * CR_;
    v8f acc;
    const float bias = b2[n];
#pragma unroll
    for (int v = 0; v < 8; ++v) acc[v] = bias;
#pragma unroll
    for (int st = 0; st < CR_ / 4; ++st) {
      const int k0 = st * 4 + koff;
      v2f b = *(const v2f*)(wb + k0);
      acc = __builtin_amdgcn_wmma_f32_16x16x4_f32(
          false, afrag[st], false, b, (short)0, acc, false, false);
    }
#pragma unroll
    for (int v = 0; v < 8; ++v)
      kerlds[(v + 8 * hi) * PITCH + n] = acc[v];
  }

  WAIT_ASYNC();          // chunk 0 resident
  __syncthreads();       // + kerlds resident

  // ---- Phase 2: involution, one 16-channel group per chunk (g == k) ------
  const int p  = tid & 15;           // pixel in tile -> coalesced stores
  const int cc = tid >> 4;           // 0..7 ; thread does channels cc, cc+8
  const int mm = mt * 16 + p;
  const int ss = mm - b0 * HW_;
  const int h  = ss / W_;
  const int w  = ss - h * W_;
  const int rb = h - h0;             // 0 or 1
  const float* kp = &kerlds[p * PITCH];

#pragma unroll 1
  for (int k = 0; k < NCHK; ++k) {
    if (k < NCHK - 1)
      stage_chunk(x, xbuf[(k + 1) & 1], b0, (k + 1) * CHCH, h0, tid);

    const float* xl = xbuf[k & 1];
    const float* kg = kp + k;        // g == k for 16-channel chunks
    const float* xA = xl + (cc * XROWS + rb) * W_;
    const float* xB = xl + ((cc + 8) * XROWS + rb) * W_;
    float accA = 0.f, accB = 0.f;
#pragma unroll
    for (int i = 0; i < K_; ++i) {
      const float* xrA = xA + i * W_;   // vertically OOB rows are zero-filled
      const float* xrB = xB + i * W_;
#pragma unroll
      for (int j = 0; j < K_; ++j) {
        const int xc = w + j - 3;
        if ((unsigned)xc < (unsigned)W_) {
          const float kv = kg[(i * K_ + j) * G_];
          accA = fmaf(xrA[xc], kv, accA);
          accB = fmaf(xrB[xc], kv, accB);
        }
      }
    }
    const size_t obase = ((size_t)b0 * C_ + k * CHCH) * HW_ + h * W_ + w;
    out[obase + (size_t)cc * HW_]       = accA;
    out[obase + (size_t)(cc + 8) * HW_] = accB;

    if (k < NCHK - 1) {
      WAIT_ASYNC();      // chunk k+1 resident
      __syncthreads();   // all waves done reading xbuf[k&1] before reuse
    }
  }
}

// ---------------------------------------------------------------------------
extern "C" void kernel_launch(void* const* d_in, const int* in_sizes, int n_in,
                              void* d_out, int out_size, void* d_ws, size_t ws_size,
                              hipStream_t stream) {
  const float* x     = (const float*)d_in[0];
  const float* w1    = (const float*)d_in[1];
  const float* b1    = (const float*)d_in[2];
  const float* gamma = (const float*)d_in[3];
  const float* beta  = (const float*)d_in[4];
  const float* w2    = (const float*)d_in[5];
  const float* b2    = (const float*)d_in[6];
  float* out = (float*)d_out;

  float* rpre = (float*)d_ws;                      // M_*CR_ floats (~12.9 MB)
  float* ab   = rpre + (size_t)M_ * CR_;           // 2*CR_ floats

  hipLaunchKernelGGL(k_reduce_gemm, dim3(M_ / 64), dim3(128), 0, stream,
                     x, w1, b1, rpre);
  hipLaunchKernelGGL(k_bnstats, dim3(CR_), dim3(256), 0, stream,
                     rpre, gamma, beta, ab);
  hipLaunchKernelGGL(k_span_involution, dim3(M_ / 16), dim3(128), 0, stream,
                     x, rpre, ab, w2, b2, out);
}